// SamplePredictor_51264729645494
// MI455X (gfx1250) — compile-verified
//
#include <hip/hip_runtime.h>
#include <math.h>

typedef __attribute__((ext_vector_type(2))) float v2f;
typedef __attribute__((ext_vector_type(8))) float v8f;

#define HID 64
#define FIN 5

// ---------------------------------------------------------------- init ------
__global__ void gcn_init(float* __restrict__ deg, float* __restrict__ emb,
                         float* __restrict__ cnt, int N, int G) {
  int t = blockIdx.x * blockDim.x + threadIdx.x;
  if (t < N) deg[t] = 1.0f;                 // self-loop baked into degree
  if (t < G * HID) emb[t] = 0.0f;
  if (t < G) cnt[t] = 0.0f;
}

// ---------------------------------------------------------------- degree ----
__global__ void gcn_degree(const int* __restrict__ dst, float* __restrict__ deg, int E) {
  int e = blockIdx.x * blockDim.x + threadIdx.x;
  if (e < E) atomicAdd(&deg[dst[e]], 1.0f);
}

__global__ void gcn_rsqrt(float* __restrict__ d, int N) {
  int t = blockIdx.x * blockDim.x + threadIdx.x;
  if (t < N) d[t] = rsqrtf(d[t]);
}

// ------------------------------------------------- layer-1 feature GEMM -----
// hs[i][f] = (x[i,:] @ W1[:,f]) * dinv[i];  acc initialized to hs (self term)
__global__ void gcn_feat1(const float* __restrict__ x, const float* __restrict__ W1,
                          const float* __restrict__ dinv, float* __restrict__ hs,
                          float* __restrict__ acc, int N) {
  int t = blockIdx.x * blockDim.x + threadIdx.x;
  if (t >= N * HID) return;
  int i = t >> 6, f = t & 63;
  float a = 0.0f;
#pragma unroll
  for (int k = 0; k < FIN; ++k) a += x[i * FIN + k] * W1[k * HID + f];
  float v = a * dinv[i];
  hs[t] = v;
  acc[t] = v;
}

// ------------------------------------------------------ edge aggregation ----
// acc[dst] += hs[src]  (per feature); one wave handles 32 consecutive feats
__global__ void gcn_edge(const float* __restrict__ hs, float* __restrict__ acc,
                         const int* __restrict__ src, const int* __restrict__ dst,
                         int E) {
  int t = blockIdx.x * blockDim.x + threadIdx.x;
  int e = t >> 6;
  if (e >= E) return;
  int f = t & 63;
  atomicAdd(acc + ((size_t)dst[e] * HID + f), hs[(size_t)src[e] * HID + f]);
}

// ------------------------------------------------------ relu(dinv*acc+b) ----
__global__ void gcn_relu(float* __restrict__ acc, const float* __restrict__ dinv,
                         const float* __restrict__ b, int N) {
  int t = blockIdx.x * blockDim.x + threadIdx.x;
  if (t >= N * HID) return;
  int i = t >> 6, f = t & 63;
  acc[t] = fmaxf(dinv[i] * acc[t] + b[f], 0.0f);
}

// ------------------------------------------------- layer-2 GEMM via WMMA ----
// hs2 = (out1 @ W2) * dinv ; acc2 initialized to hs2 (in-place over out1).
// Block: 128 threads (4 waves), 64 node rows; each wave: 16x64 output tile.
__global__ void __launch_bounds__(128) gcn_gemm2(float* __restrict__ out1_acc2,
                                                 const float* __restrict__ W2,
                                                 const float* __restrict__ dinv,
                                                 float* __restrict__ hs2, int N) {
  __shared__ float At[64 * 65];   // A tile, padded stride 65 (bank-conflict free)
  __shared__ float Bt[64 * 65];   // W2, padded
  __shared__ float dv[64];

  const int node0 = blockIdx.x * 64;

  // stage A tile + W2 into LDS with float4 coalesced loads
  for (int idx = threadIdx.x; idx < 64 * 16; idx += 128) {
    int row = idx >> 4;
    int c4 = (idx & 15) << 2;
    int node = node0 + row;
    float4 v;
    if (node < N) v = *(const float4*)(out1_acc2 + (size_t)node * HID + c4);
    else { v.x = v.y = v.z = v.w = 0.0f; }
    float* p = &At[row * 65 + c4];
    p[0] = v.x; p[1] = v.y; p[2] = v.z; p[3] = v.w;
    float4 w = *(const float4*)(W2 + row * HID + c4);
    float* q = &Bt[row * 65 + c4];
    q[0] = w.x; q[1] = w.y; q[2] = w.z; q[3] = w.w;
  }
  if (threadIdx.x < 64) {
    int node = node0 + threadIdx.x;
    dv[threadIdx.x] = (node < N) ? dinv[node] : 0.0f;
  }
  __syncthreads();

  const int lane = threadIdx.x & 31;
  const int wave = threadIdx.x >> 5;  // 0..3 -> row tile
  const int m0 = wave * 16;
  const int lo = lane & 15;
  const int hi = lane >> 4;

  v8f c0 = {}, c1 = {}, c2 = {}, c3 = {};
#pragma unroll
  for (int kb = 0; kb < 64; kb += 4) {
    const int k0 = kb + 2 * hi;
    // A 16x4 frag: lane l holds A[l&15][k0 + j]
    v2f a;
    a.x = At[(m0 + lo) * 65 + k0];
    a.y = At[(m0 + lo) * 65 + k0 + 1];
    // B 4x16 frags: VGPR r holds B[k0 + r][ntile*16 + lo]
    v2f b;
    b.x = Bt[k0 * 65 + 0 + lo];
    b.y = Bt[(k0 + 1) * 65 + 0 + lo];
    c0 = __builtin_amdgcn_wmma_f32_16x16x4_f32(false, a, false, b, (short)0, c0, false, false);
    b.x = Bt[k0 * 65 + 16 + lo];
    b.y = Bt[(k0 + 1) * 65 + 16 + lo];
    c1 = __builtin_amdgcn_wmma_f32_16x16x4_f32(false, a, false, b, (short)0, c1, false, false);
    b.x = Bt[k0 * 65 + 32 + lo];
    b.y = Bt[(k0 + 1) * 65 + 32 + lo];
    c2 = __builtin_amdgcn_wmma_f32_16x16x4_f32(false, a, false, b, (short)0, c2, false, false);
    b.x = Bt[k0 * 65 + 48 + lo];
    b.y = Bt[(k0 + 1) * 65 + 48 + lo];
    c3 = __builtin_amdgcn_wmma_f32_16x16x4_f32(false, a, false, b, (short)0, c3, false, false);
  }

  // epilogue: scale by dinv, write hs2 and init acc2 (rows are dead in out1)
  v8f cc[4] = {c0, c1, c2, c3};
#pragma unroll
  for (int nt = 0; nt < 4; ++nt) {
#pragma unroll
    for (int r = 0; r < 8; ++r) {
      int m = r + 8 * hi;             // C/D layout: VGPR r -> rows r / r+8
      int node = node0 + m0 + m;
      if (node < N) {
        float v = cc[nt][r] * dv[m0 + m];
        size_t addr = (size_t)node * HID + nt * 16 + lo;
        hs2[addr] = v;
        out1_acc2[addr] = v;
      }
    }
  }
}

// --------------------------- fused relu(dinv*acc2+b2) + mean-pool accum -----
// 1024 threads = 16 nodes x 64 feats; batch is sorted -> LDS pre-reduction
__global__ void __launch_bounds__(1024) gcn_pool(const float* __restrict__ acc2,
                                                 const float* __restrict__ dinv,
                                                 const float* __restrict__ b2,
                                                 const int* __restrict__ batch,
                                                 float* __restrict__ emb,
                                                 float* __restrict__ cnt, int N) {
  __shared__ float s[HID];
  int ty = threadIdx.x >> 6;
  int f = threadIdx.x & 63;
  int nbase = blockIdx.x * 16;
  int node = nbase + ty;
  bool valid = node < N;
  float val = 0.0f;
  int g = 0;
  if (valid) {
    val = fmaxf(dinv[node] * acc2[(size_t)node * HID + f] + b2[f], 0.0f);
    g = batch[node];
  }
  int nvalid = N - nbase;
  if (nvalid > 16) nvalid = 16;
  int gfirst = batch[nbase];
  int glast = batch[nbase + nvalid - 1];
  if (gfirst == glast) {               // block-uniform graph (common: batch sorted)
    if (threadIdx.x < HID) s[threadIdx.x] = 0.0f;
    __syncthreads();
    if (valid) atomicAdd(&s[f], val);  // ds_add_f32
    __syncthreads();
    if (threadIdx.x < HID)
      atomicAdd(&emb[(size_t)gfirst * HID + threadIdx.x], s[threadIdx.x]);
    if (threadIdx.x == 0) atomicAdd(&cnt[gfirst], (float)nvalid);
  } else {
    if (valid) {
      atomicAdd(&emb[(size_t)g * HID + f], val);
      if (f == 0) atomicAdd(&cnt[g], 1.0f);
    }
  }
}

// --------------------------------------------------------- readout MLP ------
__global__ void gcn_mlp(const float* __restrict__ emb, const float* __restrict__ cnt,
                        const float* __restrict__ Wp1, const float* __restrict__ bp1,
                        const float* __restrict__ Wp2, const float* __restrict__ bp2,
                        float* __restrict__ out, int G) {
  int g = blockIdx.x * blockDim.x + threadIdx.x;
  if (g >= G) return;
  float ic = 1.0f / fmaxf(cnt[g], 1.0f);
  float e[HID];
#pragma unroll
  for (int f = 0; f < HID; ++f) e[f] = emb[(size_t)g * HID + f] * ic;
  float h[32];
#pragma unroll
  for (int j = 0; j < 32; ++j) {
    float a = bp1[j];
    for (int f = 0; f < HID; ++f) a += e[f] * Wp1[f * 32 + j];
    h[j] = fmaxf(a, 0.0f);
  }
#pragma unroll
  for (int c = 0; c < 2; ++c) {
    float a = bp2[c];
    for (int j = 0; j < 32; ++j) a += h[j] * Wp2[j * 2 + c];
    out[g * 2 + c] = 2.0f + 3.0f / (1.0f + expf(-a));
  }
}

// ---------------------------------------------------------------------------
extern "C" void kernel_launch(void* const* d_in, const int* in_sizes, int n_in,
                              void* d_out, int out_size, void* d_ws, size_t ws_size,
                              hipStream_t stream) {
  const float* x   = (const float*)d_in[0];
  const int*   ei  = (const int*)d_in[1];
  const int*   bat = (const int*)d_in[2];
  const float* W1  = (const float*)d_in[3];
  const float* b1  = (const float*)d_in[4];
  const float* W2  = (const float*)d_in[5];
  const float* b2  = (const float*)d_in[6];
  const float* Wp1 = (const float*)d_in[7];
  const float* bp1 = (const float*)d_in[8];
  const float* Wp2 = (const float*)d_in[9];
  const float* bp2 = (const float*)d_in[10];
  float* out = (float*)d_out;

  const int N = in_sizes[0] / FIN;
  const int E = in_sizes[1] / 2;
  const int G = out_size / 2;
  const int* src = ei;
  const int* dst = ei + E;

  char* ws = (char*)d_ws;
  size_t off = 0;
  auto alloc = [&](size_t bytes) {
    char* p = ws + off;
    off += (bytes + 255) & ~(size_t)255;
    return p;
  };
  float* dinv = (float*)alloc((size_t)N * 4);          // degree -> dinv
  float* bufA = (float*)alloc((size_t)N * HID * 4);    // hs / hs2
  float* bufB = (float*)alloc((size_t)N * HID * 4);    // acc / out1 / acc2
  float* emb  = (float*)alloc((size_t)G * HID * 4);
  float* cnt  = (float*)alloc((size_t)G * 4);
  (void)ws_size; (void)n_in;

  int initCov = N;
  if (G * HID > initCov) initCov = G * HID;

  gcn_init  <<<(initCov + 255) / 256, 256, 0, stream>>>(dinv, emb, cnt, N, G);
  gcn_degree<<<(E + 255) / 256, 256, 0, stream>>>(dst, dinv, E);
  gcn_rsqrt <<<(N + 255) / 256, 256, 0, stream>>>(dinv, N);

  // layer 1
  gcn_feat1 <<<((size_t)N * HID + 255) / 256, 256, 0, stream>>>(x, W1, dinv, bufA, bufB, N);
  gcn_edge  <<<(unsigned)(((size_t)E * HID + 255) / 256), 256, 0, stream>>>(bufA, bufB, src, dst, E);
  gcn_relu  <<<(N * HID + 255) / 256, 256, 0, stream>>>(bufB, dinv, b1, N);

  // layer 2 (WMMA GEMM, writes hs2 to bufA and inits acc2 in bufB in-place)
  gcn_gemm2 <<<(N + 63) / 64, 128, 0, stream>>>(bufB, W2, dinv, bufA, N);
  gcn_edge  <<<(unsigned)(((size_t)E * HID + 255) / 256), 256, 0, stream>>>(bufA, bufB, src, dst, E);

  // fused relu + pooling, then readout MLP
  gcn_pool  <<<(N + 15) / 16, 1024, 0, stream>>>(bufB, dinv, b2, bat, emb, cnt, N);
  gcn_mlp   <<<(G + 255) / 256, 256, 0, stream>>>(emb, cnt, Wp1, bp1, Wp2, bp2, out, G);
}